// STDPLearner_90314572300877
// MI455X (gfx1250) — compile-verified
//
#include <hip/hip_runtime.h>

typedef __attribute__((ext_vector_type(2))) float v2f;
typedef __attribute__((ext_vector_type(4))) float f4;
typedef __attribute__((ext_vector_type(4))) int   i4;
typedef __attribute__((ext_vector_type(8))) float v8f;

#define VOCAB 128000
#define Bsz 1024
#define Ssz 8192

constexpr float LR_PLUS = 0.01f;
constexpr float DECAY_W = 0.99f;
constexpr float TD   = 0.81873075307798185867f;  // exp(-1/5)
constexpr float TD9  = 0.16529888822158653829f;  // exp(-9/5)
constexpr float TD16 = 0.04076220397836621204f;  // exp(-16/5)

constexpr int SEG  = 256;               // timesteps per independent segment
constexpr int NSEG = Ssz / SEG;         // 32
constexpr int HALO_CHUNKS = 12;         // 192 warm-up steps: d^192 ~ 2e-17 << fp32 ulp
constexpr int NSTRIPE = Bsz / 16;       // 64 stripes of 16 rows
constexpr int MAIN_CHUNKS = SEG / 16;   // 16
constexpr int PF = 64;                  // prefetch distance: 4 chunks (one 256B span)

// One wave per workgroup, 2048 independent blocks: maximizes concurrent waves
// (in-flight bytes) for a latency-limited stream. Fully LDS-free: every value a
// lane needs lives in its own row, so tiles stay in registers and chunk c+1's
// loads are issued before chunk c is consumed (register double-buffering).
__global__ __launch_bounds__(32) void stdp_scan_scatter(
    const int* __restrict__ tok, const float* __restrict__ spk,
    float* __restrict__ grad)
{
  const int lane = threadIdx.x;         // 0..31
  const bool hi  = lane >= 16;
  const int  m   = lane & 15;           // A-row (i) / B & C column (n) index

  const int seg  = blockIdx.x / NSTRIPE;            // wave-uniform
  const int row0 = (blockIdx.x % NSTRIPE) * 16;
  const int t0   = seg * SEG;

  // Per-lane row pointers: lanes m and m+16 read the same spike cachelines
  // (merged by the coalescer); token reads are split by half-wave (ibase).
  const float* gs = spk + (size_t)(row0 + m) * Ssz;
  const int*   gt = tok + (size_t)(row0 + m) * Ssz + (hi ? 8 : 0);

  // A fragments: A[i][j] = d^(i-j) for j<=i (lower-triangular decay matrix).
  // 16x4 f32 A layout: VGPR0 K = kb (lanes 0-15) / kb+2 (lanes 16-31),
  // VGPR1 K = kb+1 / kb+3, M = lane&15.
  v2f afrag[4];
#pragma unroll
  for (int kk = 0; kk < 4; ++kk) {
    const int kb = 4 * kk + (hi ? 2 : 0);
    afrag[kk].x = (kb     <= m) ? __expf(-0.2f * (float)(m - kb))     : 0.f;
    afrag[kk].y = (kb + 1 <= m) ? __expf(-0.2f * (float)(m - kb - 1)) : 0.f;
  }

  float carry = 0.f;  // trace value at the step before the current chunk (row n)
  const int c0 = (seg == 0) ? 0 : -HALO_CHUNKS;     // wave-uniform

  // Stage 0: loads for the first chunk.
  f4 cs[4]; i4 ct[2];
  {
    const float* g = gs + (t0 + c0 * 16);
#pragma unroll
    for (int q = 0; q < 4; ++q) cs[q] = *(const f4*)(g + 4 * q);
    if (c0 >= 0) {
      const int* h = gt + (t0 + c0 * 16);
      ct[0] = __builtin_nontemporal_load((const i4*)h);
      ct[1] = __builtin_nontemporal_load((const i4*)(h + 4));
    }
  }

  // Halo chunks (c<0) run the same WMMA body to advance the carry only.
  for (int c = c0; c < MAIN_CHUNKS; ++c) {
    const int tb = t0 + c * 16;
    const bool has_next = (c + 1 < MAIN_CHUNKS);      // wave-uniform

    // Issue next chunk's loads before consuming this one (hide load latency).
    f4 ns[4]; i4 nt[2];
    if (has_next) {
      const float* g = gs + (tb + 16);
#pragma unroll
      for (int q = 0; q < 4; ++q) ns[q] = *(const f4*)(g + 4 * q);
      if (c + 1 >= 0) {
        const int* h = gt + (tb + 16);
        nt[0] = __builtin_nontemporal_load((const i4*)h);
        nt[1] = __builtin_nontemporal_load((const i4*)(h + 4));
      }
      if (tb + PF < Ssz) {                            // pre-stage L2, 4 chunks out
        __builtin_prefetch(gs + (tb + PF), 0, 1);
        __builtin_prefetch(gt + (tb + PF), 0, 1);
      }
    }

    // local[i][n] = sum_{j<=i} d^(i-j)*s[j][n] via 4 chained 16x16x4 f32 WMMAs.
    // B layout: VGPR0 K = kb (lanes 0-15) / kb+2 (lanes 16-31), VGPR1 K = kb+1
    // / kb+3, N = lane&15 -> one cndmask per element from the lane's own row.
    v8f acc = {0.f, 0.f, 0.f, 0.f, 0.f, 0.f, 0.f, 0.f};
#pragma unroll
    for (int kk = 0; kk < 4; ++kk) {
      v2f b;
      b.x = hi ? cs[kk][2] : cs[kk][0];
      b.y = hi ? cs[kk][3] : cs[kk][1];
      acc = __builtin_amdgcn_wmma_f32_16x16x4_f32(
          /*neg_a=*/false, afrag[kk], /*neg_b=*/false, b,
          /*c_mod=*/(short)0, acc, /*reuse_a=*/false, /*reuse_b=*/false);
    }

    if (c >= 0) {
      // C/D layout: VGPR r -> (i = r + (hi?8:0), n = lane&15).
      float dpw = hi ? TD9 : TD;        // d^(i+1) at r = 0
#pragma unroll
      for (int r = 0; r < 8; ++r) {
        const int q = r >> 2, e = r & 3;
        const float sv = hi ? cs[2 + q][e] : cs[q][e];
        const int   tk = ct[q][e];
        const float full = acc[r] + dpw * carry;   // trace[i] = local + d^(i+1)*carry
        dpw *= TD;
        if (sv != 0.f)                  // spikes are exactly {0,1}
          atomicAdd(&grad[tk], LR_PLUS * full);
      }
    }
    // carry' = trace at i=15 = local[15] + d^16*carry; hi lane 16+n holds row n
    carry = __shfl(acc[7] + TD16 * carry, 16 + m, 32);

    if (has_next) {
#pragma unroll
      for (int q = 0; q < 4; ++q) cs[q] = ns[q];
      if (c + 1 >= 0) { ct[0] = nt[0]; ct[1] = nt[1]; }
    }
  }
}

__global__ void zero_f4(f4* __restrict__ p, int n4) {
  const int i = blockIdx.x * blockDim.x + threadIdx.x;
  if (i < n4) p[i] = (f4){0.f, 0.f, 0.f, 0.f};
}

__global__ void finalize_w(const f4* __restrict__ w, const f4* __restrict__ grad,
                           f4* __restrict__ out, int n4) {
  const int i = blockIdx.x * blockDim.x + threadIdx.x;
  if (i < n4) {
    const f4 v = (w[i] + grad[i]) * DECAY_W;
    f4 r;
#pragma unroll
    for (int k = 0; k < 4; ++k) r[k] = fminf(fmaxf(v[k], 0.f), 1.f);
    out[i] = r;
  }
}

extern "C" void kernel_launch(void* const* d_in, const int* in_sizes, int n_in,
                              void* d_out, int out_size, void* d_ws, size_t ws_size,
                              hipStream_t stream) {
  (void)in_sizes; (void)n_in; (void)out_size; (void)ws_size;
  const int*   tok = (const int*)d_in[0];
  const float* spk = (const float*)d_in[1];
  const float* w   = (const float*)d_in[2];
  float* out  = (float*)d_out;
  float* grad = (float*)d_ws;   // VOCAB floats of scratch (512 KB), re-zeroed per call

  constexpr int N4 = VOCAB / 4;         // 32000
  zero_f4<<<(N4 + 255) / 256, 256, 0, stream>>>((f4*)grad, N4);
  stdp_scan_scatter<<<NSEG * NSTRIPE, 32, 0, stream>>>(tok, spk, grad);
  finalize_w<<<(N4 + 255) / 256, 256, 0, stream>>>(
      (const f4*)w, (const f4*)grad, (f4*)out, N4);
}